// RotateImage_63067299775248
// MI455X (gfx1250) — compile-verified
//
#include <hip/hip_runtime.h>
#include <math.h>

#define NB 64
#define NC 3
#define NH 512
#define NW 512
#define ROWS_PER_BLOCK 2   // 256 threads = 2 rows x 128 threads, 4 px/thread

// ---------------------------------------------------------------------------
// Kernel 1: precompute {cos(theta), sin(theta)} per batch into d_ws.
// One wave-pair of work total; removes the big scalarized sincos (Payne-Hanek
// path and all) from the 49k-block hot kernel.
// ---------------------------------------------------------------------------
__global__ __launch_bounds__(64) void precompute_trig_kernel(
    const float* __restrict__ angles,  // (B,)
    float2* __restrict__ cs)           // (B,) {cos, sin}
{
    const int b = threadIdx.x;
    if (b < NB) {
        const float theta = angles[b] * 0.017453292519943295f; // deg2rad
        float s, c;
        sincosf(theta, &s, &c);
        cs[b] = make_float2(c, s);
    }
}

// ---------------------------------------------------------------------------
// Kernel 2: bandwidth-bound nearest-neighbor rotation.
// ---------------------------------------------------------------------------
__global__ __launch_bounds__(256) void rotate_nearest_kernel(
    const float* __restrict__ x,       // (B, C, H, W)
    const float2* __restrict__ cs,     // (B,) {cos, sin}
    float* __restrict__ out)           // (B, C, H, W)
{
    // Block -> (slice = b*C + ch, pair of rows). blockIdx-derived => uniform.
    const int slice = blockIdx.x / (NH / ROWS_PER_BLOCK);
    const int row0  = (blockIdx.x % (NH / ROWS_PER_BLOCK)) * ROWS_PER_BLOCK;
    const int b     = slice / NC;

    const int tid  = threadIdx.x;
    const int row  = row0 + (tid >> 7);   // 128 threads per row
    const int col0 = (tid & 127) << 2;    // 4 consecutive pixels per thread

    // Uniform per block -> single s_load_b64.
    const float2 t = cs[b];
    const float c = t.x;
    const float s = t.y;

    const float cy = (NH - 1) * 0.5f;
    const float cx = (NW - 1) * 0.5f;
    const float Yf = (float)row - cy;

    const float* __restrict__ src = x + (size_t)slice * (NH * NW);

    float4 o;
    float* op = &o.x;

    // L2 prefetch a few rows ahead of the slanted gather band (address
    // clamped in-bounds: WGP-scope prefetch is non-speculative on gfx1250).
    {
        const float Xf = (float)col0 - cx;
        const float sy = s * Xf + c * Yf + cy;
        const float sx = c * Xf - s * Yf + cx;
        int iyp = __float2int_rn(sy);
        int ixp = __float2int_rn(sx);
        iyp = min(max(iyp + 4, 0), NH - 1);
        ixp = min(max(ixp, 0), NW - 1);
        __builtin_prefetch(src + (size_t)iyp * NW + ixp, 0, 3); // global_prefetch_b8
    }

#pragma unroll
    for (int j = 0; j < 4; ++j) {
        const float Xf = (float)(col0 + j) - cx;
        const float sx = c * Xf - s * Yf + cx;
        const float sy = s * Xf + c * Yf + cy;

        // jnp.round == round-half-to-even -> v_cvt_i32_f32 with RNE
        const int ix = __float2int_rn(sx);
        const int iy = __float2int_rn(sy);

        const bool valid = (ix >= 0) & (ix < NW) & (iy >= 0) & (iy < NH);

        const int ixc = min(max(ix, 0), NW - 1);
        const int iyc = min(max(iy, 0), NH - 1);

        const float v = src[(size_t)iyc * NW + ixc]; // gather (global_load_b32)
        op[j] = valid ? v : 0.0f;
    }

    // Coalesced 16B store: half-wave covers 256 contiguous bytes.
    float4* outp = (float4*)(out + (size_t)slice * (NH * NW) + (size_t)row * NW + col0);
    *outp = o;
}

extern "C" void kernel_launch(void* const* d_in, const int* in_sizes, int n_in,
                              void* d_out, int out_size, void* d_ws, size_t ws_size,
                              hipStream_t stream) {
    (void)in_sizes; (void)n_in; (void)out_size; (void)ws_size;
    const float* x      = (const float*)d_in[0];
    const float* angles = (const float*)d_in[1];
    float* out          = (float*)d_out;
    float2* cs          = (float2*)d_ws;   // 64 * 8 B = 512 B of workspace

    precompute_trig_kernel<<<1, 64, 0, stream>>>(angles, cs);

    const int blocks = NB * NC * (NH / ROWS_PER_BLOCK); // 49152
    rotate_nearest_kernel<<<blocks, 256, 0, stream>>>(x, cs, out);
}